// RGCN_DDI_Model_23441931502093
// MI455X (gfx1250) — compile-verified
//
#include <hip/hip_runtime.h>

// ============================================================================
// RGCN-DDI on MI455X (gfx1250).  All matmuls in fp32 WMMA (16x16x4) — the
// workload is ~27 GFLOP total and entirely data-movement bound, so fp32 is
// the right precision; the win is keeping gather/scatter in the 192 MB L2.
//
//   1) zero cnt/header; block-LDS histogram -> cnt[(dst,rel)], relCount[rel]
//   2) 32-bin scan -> relOffset[], tileBase[] (128-edge super-tiles/relation)
//   3) two-level counting-sort scatter of edge ids into perm[] (one global
//      atomic per bin per block, never per edge -> no hot-line serialization)
//   4) per layer: zero agg; message kernel (8 waves/block, one 128-edge
//      super-tile of a single relation, W[r] staged once in LDS; per wave a
//      16x64 WMMA GEMM on rows gathered from L2-resident x; scale by
//      1/cnt and cacheline-coalesced global_atomic_add_f32 into agg);
//      root kernel (WMMA x@root from LDS + agg + bias, ReLU layer 1)
//   5) fused classifier: gather concat(x2[d1],x2[d2]) 16x128 tiles,
//      WMMA @ cW1 (LDS) -> ReLU -> LDS -> WMMA @ cW2 (LDS) -> out [B,32]
// ============================================================================

#define R_REL 32
#define H_DIM 64

typedef float v2f __attribute__((ext_vector_type(2)));
typedef float v8f __attribute__((ext_vector_type(8)));

__device__ __forceinline__ v8f wmma_k4(v2f a, v2f b, v8f c) {
  // D = A(16x4, f32) * B(4x16, f32) + C(16x16, f32)
  return __builtin_amdgcn_wmma_f32_16x16x4_f32(
      /*neg_a=*/false, a, /*neg_b=*/false, b,
      /*c_mod=*/(short)0, c, /*reuse_a=*/false, /*reuse_b=*/false);
}

__device__ __forceinline__ void atomic_add_f32(float* p, float v) {
  __hip_atomic_fetch_add(p, v, __ATOMIC_RELAXED, __HIP_MEMORY_SCOPE_AGENT);
}

// ---------------------------------------------------------------------------
__global__ void k_zero_u32(unsigned* __restrict__ p, long n) {
  long i = (long)blockIdx.x * blockDim.x + threadIdx.x;
  if (i < n) p[i] = 0u;
}

// header layout (u32): [0..31]=relCount [32..63]=relCursor
//                      [64..96]=relOffset [97..129]=tileBase (128-edge units)
__global__ __launch_bounds__(256) void k_hist(
    const int* __restrict__ dst, const int* __restrict__ et,
    unsigned* __restrict__ cnt, unsigned* __restrict__ header, long E) {
  __shared__ unsigned h[R_REL];
  int tid = threadIdx.x;
  if (tid < R_REL) h[tid] = 0u;
  __syncthreads();
  long e = (long)blockIdx.x * blockDim.x + tid;
  if (e < E) {
    int t = et[e];
    atomicAdd(&cnt[(long)dst[e] * R_REL + t], 1u);  // 1.6M distinct addrs: fine
    atomicAdd(&h[t], 1u);                           // LDS: 8/bin/block
  }
  __syncthreads();
  if (tid < R_REL && h[tid]) atomicAdd(&header[tid], h[tid]);  // 1 per bin/block
}

__global__ void k_scan(unsigned* __restrict__ header) {
  if (threadIdx.x == 0 && blockIdx.x == 0) {
    unsigned ro = 0, tb = 0;
    for (int r = 0; r < R_REL; ++r) {
      header[64 + r] = ro;
      header[97 + r] = tb;
      ro += header[r];
      tb += (header[r] + 127u) >> 7;   // super-tiles of 128 edges
    }
    header[64 + R_REL] = ro;
    header[97 + R_REL] = tb;
  }
}

// two-level counting sort: per-block LDS ranks + one cursor atomic per bin
__global__ __launch_bounds__(256) void k_scatter(
    const int* __restrict__ et, unsigned* __restrict__ header,
    unsigned* __restrict__ perm, long E) {
  __shared__ unsigned hcnt[R_REL];
  __shared__ unsigned hbase[R_REL];
  int tid = threadIdx.x;
  if (tid < R_REL) hcnt[tid] = 0u;
  __syncthreads();
  long e = (long)blockIdx.x * blockDim.x + tid;
  bool ok = e < E;
  int t = 0;
  unsigned rank = 0;
  if (ok) {
    t = et[e];
    rank = atomicAdd(&hcnt[t], 1u);
  }
  __syncthreads();
  if (tid < R_REL) {
    unsigned c = hcnt[tid];
    hbase[tid] = c ? atomicAdd(&header[32 + tid], c) : 0u;
  }
  __syncthreads();
  if (ok) {
    unsigned pos = header[64 + t] + hbase[t] + rank;
    perm[pos] = (unsigned)e;
  }
}

// ---------------------------------------------------------------------------
// Message kernel: 8 waves/block; one 128-edge super-tile of ONE relation.
// W[r] staged once in LDS; each wave does a 16x64 = (16x64)@(64x64) WMMA GEMM.
__global__ __launch_bounds__(256) void k_msg(
    const float* __restrict__ x, const float* __restrict__ W,
    const int* __restrict__ src, const int* __restrict__ dst,
    const unsigned* __restrict__ perm, const unsigned* __restrict__ cnt,
    const unsigned* __restrict__ header, float* __restrict__ agg) {
  __shared__ float Blds[H_DIM * H_DIM];  // 16 KB: W[r]
  __shared__ int   sSrc[128];
  __shared__ int   sDst[128];
  __shared__ float sNorm[128];

  unsigned t = blockIdx.x;
  const unsigned* tileBase = header + 97;
  if (t >= tileBase[R_REL]) return;       // block-uniform

  int r = 0;
  while (tileBase[r + 1] <= t) ++r;       // <=32 scalar iterations, uniform
  unsigned baseInRel = (t - tileBase[r]) * 128u;
  int count = (int)min(128u, header[r] - baseInRel);
  unsigned segStart = header[64 + r] + baseInRel;

  int tid = threadIdx.x;
  {  // stage W[r]: 4096 floats, 4 float4 per thread, fully coalesced
    const float4* w4 = (const float4*)(W + (long)r * H_DIM * H_DIM);
    float4* b4 = (float4*)Blds;
#pragma unroll
    for (int i = tid; i < H_DIM * H_DIM / 4; i += 256) b4[i] = w4[i];
  }
  if (tid < 128) {  // per-edge metadata for the super-tile
    int si = 0, di = 0;
    float nm = 0.0f;
    if (tid < count) {
      int e = (int)perm[segStart + tid];
      si = src[e];
      di = dst[e];
      unsigned c = cnt[(long)di * R_REL + r];
      nm = 1.0f / (float)max(c, 1u);
    }
    sSrc[tid] = si; sDst[tid] = di; sNorm[tid] = nm;
  }
  __syncthreads();

  int wave = tid >> 5, lane = tid & 31;
  int row = lane & 15, half = lane >> 4;
  int myBase = wave * 16;
  int myCount = min(16, count - myBase);
  if (myCount <= 0) return;               // wave-uniform; no barriers follow

  // A fragments: lane's half-row (16 float2 = 32 VGPRs), reused for 4 slabs
  const float* arow = x + (long)sSrc[myBase + row] * H_DIM + half * 2;
  v2f afrag[16];
#pragma unroll
  for (int s = 0; s < 16; ++s) afrag[s] = *(const v2f*)(arow + 4 * s);

#pragma unroll
  for (int ns = 0; ns < 4; ++ns) {
    int col = ns * 16 + row;
    v8f c = {0.f, 0.f, 0.f, 0.f, 0.f, 0.f, 0.f, 0.f};
#pragma unroll
    for (int s = 0; s < 16; ++s) {
      int kb = 4 * s + half * 2;
      v2f b;
      b.x = Blds[kb * H_DIM + col];
      b.y = Blds[(kb + 1) * H_DIM + col];
      c = wmma_k4(afrag[s], b, c);
    }
#pragma unroll
    for (int j = 0; j < 8; ++j) {
      int m = half * 8 + j;
      if (m < myCount) {
        float v = c[j] * sNorm[myBase + m];
        // lanes 0..15 cover 16 consecutive floats -> one 64B line per wave op
        atomic_add_f32(&agg[(long)sDst[myBase + m] * H_DIM + col], v);
      }
    }
  }
}

// ---------------------------------------------------------------------------
// Root kernel: out[n] = act(agg[n] + x[n] @ root + bias).  4 waves/block.
__global__ __launch_bounds__(128) void k_root(
    const float* __restrict__ x, const float* __restrict__ root,
    const float* __restrict__ bias, const float* __restrict__ agg,
    float* __restrict__ out, int nrows, int do_relu) {
  __shared__ float B[H_DIM * H_DIM];
  int tid = threadIdx.x;
  {
    const float4* r4 = (const float4*)root;
    float4* b4 = (float4*)B;
#pragma unroll
    for (int i = tid; i < H_DIM * H_DIM / 4; i += 128) b4[i] = r4[i];
  }
  __syncthreads();

  int wave = tid >> 5, lane = tid & 31;
  int row = lane & 15, half = lane >> 4;
  long rowBase = (long)blockIdx.x * 64 + wave * 16;
  int count = (int)min((long)16, (long)nrows - rowBase);
  if (count <= 0) return;                 // after the only barrier: safe

  long ldRow = rowBase + min(row, count - 1);   // clamp for gather
  const float* arow = x + ldRow * H_DIM + half * 2;
  v2f afrag[16];
#pragma unroll
  for (int s = 0; s < 16; ++s) afrag[s] = *(const v2f*)(arow + 4 * s);

#pragma unroll
  for (int ns = 0; ns < 4; ++ns) {
    int col = ns * 16 + row;
    float bc = bias[col];
    v8f c = {0.f, 0.f, 0.f, 0.f, 0.f, 0.f, 0.f, 0.f};
#pragma unroll
    for (int s = 0; s < 16; ++s) {
      int kb = 4 * s + half * 2;
      v2f b;
      b.x = B[kb * H_DIM + col];
      b.y = B[(kb + 1) * H_DIM + col];
      c = wmma_k4(afrag[s], b, c);
    }
#pragma unroll
    for (int j = 0; j < 8; ++j) {
      int m = half * 8 + j;
      if (m < count) {
        long g = rowBase + m;
        float v = c[j] + agg[g * H_DIM + col] + bc;
        if (do_relu) v = fmaxf(v, 0.0f);
        out[g * H_DIM + col] = v;
      }
    }
  }
}

// ---------------------------------------------------------------------------
// Fused classifier: h=concat(x2[d1],x2[d2]); t=relu(h@cW1+cb1); out=t@cW2+cb2
// 4 waves/block, 64 pairs/block.  LDS: 32K (cW1) + 8K (cW2) + 16K (t) = 56 KB.
__global__ __launch_bounds__(128) void k_cls(
    const float* __restrict__ x2, const int* __restrict__ d1,
    const int* __restrict__ d2, const float* __restrict__ cW1,
    const float* __restrict__ cb1, const float* __restrict__ cW2,
    const float* __restrict__ cb2, float* __restrict__ out, int B) {
  __shared__ float B1[2 * H_DIM * H_DIM];   // 128x64
  __shared__ float B2[H_DIM * R_REL];       // 64x32
  __shared__ float T[4][16 * H_DIM];        // per-wave 16x64 hidden tile
  int tid = threadIdx.x;
  {
    const float4* s4 = (const float4*)cW1;
    float4* q4 = (float4*)B1;
#pragma unroll
    for (int i = tid; i < 2 * H_DIM * H_DIM / 4; i += 128) q4[i] = s4[i];
    const float4* s2 = (const float4*)cW2;
    float4* q2 = (float4*)B2;
#pragma unroll
    for (int i = tid; i < H_DIM * R_REL / 4; i += 128) q2[i] = s2[i];
  }
  __syncthreads();

  int wave = tid >> 5, lane = tid & 31;
  int row = lane & 15, half = lane >> 4;
  long pairBase = (long)blockIdx.x * 64 + wave * 16;
  int count = (int)min((long)16, (long)B - pairBase);
  bool active = count > 0;

  int p = active ? (int)(pairBase + min(row, count - 1)) : 0;
  const float* ra = x2 + (long)d1[p] * H_DIM;
  const float* rb = x2 + (long)d2[p] * H_DIM;

  // A over K=128: 32 float2's per lane (64 VGPRs)
  v2f afrag[32];
#pragma unroll
  for (int s = 0; s < 32; ++s) {
    int kb = 4 * s + half * 2;
    const float* srcp = (kb < H_DIM) ? (ra + kb) : (rb + kb - H_DIM);
    afrag[s] = *(const v2f*)srcp;
  }

  float* t = T[wave];
#pragma unroll
  for (int ns = 0; ns < 4; ++ns) {
    int col = ns * 16 + row;
    v8f c = {0.f, 0.f, 0.f, 0.f, 0.f, 0.f, 0.f, 0.f};
#pragma unroll
    for (int s = 0; s < 32; ++s) {
      int kb = 4 * s + half * 2;
      v2f b;
      b.x = B1[kb * H_DIM + col];
      b.y = B1[(kb + 1) * H_DIM + col];
      c = wmma_k4(afrag[s], b, c);
    }
    float bc = cb1[col];
#pragma unroll
    for (int j = 0; j < 8; ++j) {
      int m = half * 8 + j;
      t[m * H_DIM + col] = fmaxf(c[j] + bc, 0.0f);
    }
  }
  __syncthreads();   // all waves reach this barrier (no early returns above)

  v2f a2[16];
#pragma unroll
  for (int s = 0; s < 16; ++s) {
    int kb = 4 * s + half * 2;
    a2[s] = *(const v2f*)(t + row * H_DIM + kb);
  }

#pragma unroll
  for (int ns = 0; ns < 2; ++ns) {
    int col = ns * 16 + row;
    v8f c = {0.f, 0.f, 0.f, 0.f, 0.f, 0.f, 0.f, 0.f};
#pragma unroll
    for (int s = 0; s < 16; ++s) {
      int kb = 4 * s + half * 2;
      v2f b;
      b.x = B2[kb * R_REL + col];
      b.y = B2[(kb + 1) * R_REL + col];
      c = wmma_k4(a2[s], b, c);
    }
    float bc = cb2[col];
#pragma unroll
    for (int j = 0; j < 8; ++j) {
      int m = half * 8 + j;
      if (active && m < count)
        out[(pairBase + m) * R_REL + col] = c[j] + bc;
    }
  }
}

// ---------------------------------------------------------------------------
extern "C" void kernel_launch(void* const* d_in, const int* in_sizes, int n_in,
                              void* d_out, int out_size, void* d_ws, size_t ws_size,
                              hipStream_t stream) {
  const int*   edge_index = (const int*)d_in[0];
  const int*   edge_type  = (const int*)d_in[1];
  const int*   drug1      = (const int*)d_in[2];
  const int*   drug2      = (const int*)d_in[3];
  const float* emb        = (const float*)d_in[4];
  const float* W1         = (const float*)d_in[5];
  const float* root1      = (const float*)d_in[6];
  const float* b1         = (const float*)d_in[7];
  const float* W2         = (const float*)d_in[8];
  const float* root2      = (const float*)d_in[9];
  const float* b2         = (const float*)d_in[10];
  const float* cW1        = (const float*)d_in[11];
  const float* cb1        = (const float*)d_in[12];
  const float* cW2        = (const float*)d_in[13];
  const float* cb2        = (const float*)d_in[14];

  const long E = in_sizes[1];
  const long N = in_sizes[4] / H_DIM;
  const int  B = in_sizes[2];
  const int* src = edge_index;
  const int* dst = edge_index + E;

  // workspace layout (~51 MB total)
  char* ws = (char*)d_ws;
  size_t off = 0;
  auto take = [&](size_t bytes) {
    size_t cur = off;
    off = (off + bytes + 255) & ~(size_t)255;
    return cur;
  };
  unsigned* cnt    = (unsigned*)(ws + take((size_t)N * R_REL * 4));
  unsigned* header = (unsigned*)(ws + take(256 * 4));
  unsigned* perm   = (unsigned*)(ws + take((size_t)E * 4));
  float*    aggb   = (float*)(ws + take((size_t)N * H_DIM * 4));
  float*    x1     = (float*)(ws + take((size_t)N * H_DIM * 4));
  float*    x2     = (float*)(ws + take((size_t)N * H_DIM * 4));
  (void)ws_size;

  const long nCnt = N * R_REL;
  const int  ZB = 256;
  // ---- sort / normalization setup (shared by both layers) ----
  k_zero_u32<<<(unsigned)((nCnt + 256 + ZB - 1) / ZB), ZB, 0, stream>>>(cnt, nCnt + 256);
  k_hist<<<(unsigned)((E + ZB - 1) / ZB), ZB, 0, stream>>>(dst, edge_type, cnt, header, E);
  k_scan<<<1, 32, 0, stream>>>(header);
  k_scatter<<<(unsigned)((E + ZB - 1) / ZB), ZB, 0, stream>>>(edge_type, header, perm, E);

  const unsigned maxTiles = (unsigned)(E / 128 + R_REL + 1);   // 128-edge super-tiles
  const unsigned rootGrid = (unsigned)((N + 63) / 64);
  const long nAgg = N * H_DIM;

  // ---- layer 1 ----
  k_zero_u32<<<(unsigned)((nAgg + ZB - 1) / ZB), ZB, 0, stream>>>((unsigned*)aggb, nAgg);
  k_msg<<<maxTiles, 256, 0, stream>>>(emb, W1, src, dst, perm, cnt, header, aggb);
  k_root<<<rootGrid, 128, 0, stream>>>(emb, root1, b1, aggb, x1, (int)N, 1);

  // ---- layer 2 ----
  k_zero_u32<<<(unsigned)((nAgg + ZB - 1) / ZB), ZB, 0, stream>>>((unsigned*)aggb, nAgg);
  k_msg<<<maxTiles, 256, 0, stream>>>(x1, W2, src, dst, perm, cnt, header, aggb);
  k_root<<<rootGrid, 128, 0, stream>>>(x1, root2, b2, aggb, x2, (int)N, 0);

  // ---- classifier ----
  k_cls<<<(unsigned)((B + 63) / 64), 128, 0, stream>>>(
      x2, drug1, drug2, cW1, cb1, cW2, cb2, (float*)d_out, B);
}